// GCNNOperator_53429393162749
// MI455X (gfx1250) — compile-verified
//
#include <hip/hip_runtime.h>

typedef float v2f __attribute__((ext_vector_type(2)));
typedef float v8f __attribute__((ext_vector_type(8)));

#define F 128      // F_OLD == F_NEW
#define BATCH 4

// ---------------------------------------------------------------------------
// Stage 1: per-node GEMM  h[b,n,:] = sum_k x[b,n,k] * W[n,k,:]
// One block (256 thr = 8 wave32) per node; wave w owns output cols [16w,16w+16).
// WMMA f32 16x16x4: A = x rows (M=0..3 valid, rows 4..15 zero-padded in LDS so
// every lane issues an unconditional ds_load_b64 — no exec-mask juggling).
// Software-pipelined one k-step ahead so loads overlap the WMMA.
// ---------------------------------------------------------------------------
__global__ void __launch_bounds__(256)
gcnn_gemm_wmma(const float* __restrict__ x,
               const float* __restrict__ W,
               float* __restrict__ h, int nNodes)
{
    const int n     = blockIdx.x;
    const int tid   = threadIdx.x;
    const int wv    = tid >> 5;          // wave id 0..7
    const int lane  = tid & 31;
    const int l     = lane & 15;
    const int khalf = (lane >> 4) << 1;  // 0 for lanes 0-15, 2 for lanes 16-31

    // A-tile in LDS, padded to the full 16 WMMA rows (rows 4..15 = 0). 8 KB.
    __shared__ float xs[16 * F];
    for (int i = tid; i < 16 * F; i += 256) {
        int b = i >> 7, k = i & (F - 1);
        xs[i] = (b < BATCH) ? x[((size_t)b * nNodes + n) * F + k] : 0.0f;
    }
    __syncthreads();

    const float* Wn   = W + (size_t)n * F * F;        // node weight, row-major [k][g]
    const float* xrow = xs + l * F + khalf;           // lane's A row base (K offset khalf)
    const float* Bp0  = Wn + (size_t)khalf * F + wv * 16 + l;

    v8f acc = {};

    // A-matrix 16x4 f32 layout: V0 = {K=0 | K=2}, V1 = {K=1 | K=3}; M = lane%16
    // B-matrix 4x16 f32: rows K striped the same way, N = wv*16 + lane%16
    v2f A  = *(const v2f*)xrow;                       // ds_load_b64 (k0, k0+1)
    v2f Bv = { Bp0[0], Bp0[F] };

    #pragma unroll 4
    for (int kt = 1; kt < F / 4; ++kt) {
        v2f An = *(const v2f*)(xrow + kt * 4);
        const float* Bpn = Bp0 + (size_t)kt * 4 * F;
        v2f Bn = { Bpn[0], Bpn[F] };
        // 8 args: (neg_a, A, neg_b, B, c_mod, C, reuse_a, reuse_b)
        acc = __builtin_amdgcn_wmma_f32_16x16x4_f32(
            false, A, false, Bv, (short)0, acc, false, false);
        A = An; Bv = Bn;
    }
    acc = __builtin_amdgcn_wmma_f32_16x16x4_f32(
        false, A, false, Bv, (short)0, acc, false, false);

    // C/D layout: VGPR r -> M=r (lanes 0-15), M=r+8 (lanes 16-31). Batch = M 0..3.
    if (lane < 16) {
        const int g = wv * 16 + lane;
        #pragma unroll
        for (int r = 0; r < BATCH; ++r)
            h[((size_t)r * nNodes + n) * F + g] = acc[r];
    }
}

// ---------------------------------------------------------------------------
// Stage 2a: out = bias (broadcast over batch); float4-vectorized.
// ---------------------------------------------------------------------------
__global__ void __launch_bounds__(256)
gcnn_bias_init(const float4* __restrict__ bias4, float4* __restrict__ out4,
               int total4, int nf4)
{
    int i = blockIdx.x * blockDim.x + threadIdx.x;
    if (i < total4) out4[i] = bias4[i % nf4];
}

// ---------------------------------------------------------------------------
// Stage 2b: edge scatter  out[:, src, :] += w_e * h[:, dst, :]
// One block per edge, 128 lanes = one feature row, loop over batch.
// h (20.5 MB) is L2-resident; fp32 atomics resolve in L2.
// ---------------------------------------------------------------------------
__global__ void __launch_bounds__(128)
gcnn_scatter(const float* __restrict__ h,
             const int*   __restrict__ esrc,
             const int*   __restrict__ edst,
             const float* __restrict__ ew,
             float* __restrict__ out, int nNodes, int nEdges)
{
    const int e = blockIdx.x;
    if (e >= nEdges) return;
    const int   s = esrc[e];
    const int   d = edst[e];
    const float w = ew[e];
    const int   g = threadIdx.x;

    #pragma unroll
    for (int b = 0; b < BATCH; ++b) {
        float v = w * h[((size_t)b * nNodes + d) * F + g];
        unsafeAtomicAdd(&out[((size_t)b * nNodes + s) * F + g], v);  // global_atomic_add_f32
    }
}

// ---------------------------------------------------------------------------
extern "C" void kernel_launch(void* const* d_in, const int* in_sizes, int n_in,
                              void* d_out, int out_size, void* d_ws, size_t ws_size,
                              hipStream_t stream)
{
    const float* x    = (const float*)d_in[0];  // [B,N,F]
    const float* W    = (const float*)d_in[1];  // [N,F,F]
    const float* bias = (const float*)d_in[2];  // [N,F]
    const int*   esrc = (const int*)  d_in[3];  // [E]
    const int*   edst = (const int*)  d_in[4];  // [E]
    const float* ew   = (const float*)d_in[5];  // [E]
    float* out = (float*)d_out;                 // [B,N,F]

    const int nNodes = in_sizes[2] / F;
    const int nEdges = in_sizes[3];

    float* h = (float*)d_ws;                    // [B,N,F] intermediate (20.97 MB)

    gcnn_gemm_wmma<<<nNodes, 256, 0, stream>>>(x, W, h, nNodes);

    const int total4 = (BATCH * nNodes * F) / 4;
    const int nf4    = (nNodes * F) / 4;
    gcnn_bias_init<<<(total4 + 255) / 256, 256, 0, stream>>>(
        (const float4*)bias, (float4*)out, total4, nf4);

    gcnn_scatter<<<nEdges, 128, 0, stream>>>(h, esrc, edst, ew, out, nNodes, nEdges);
}